// Grapher_39470749450307
// MI455X (gfx1250) — compile-verified
//
#include <hip/hip_runtime.h>
#include <hip/hip_bf16.h>

// ---------------------------------------------------------------------------
// Sizes fixed by the reference problem
// ---------------------------------------------------------------------------
#define BB   128      // batch
#define SS   1024     // sequence
#define HH   768      // hidden
#define CC   150      // mlp width / classes
#define CP   160      // CC padded: multiple of 16 (N tiles) and of 4 (K step)
#define NN   8        // MAX_NODES
#define SEP  3        // NODE_SEP_ID

typedef __attribute__((ext_vector_type(2))) float v2f;
typedef __attribute__((ext_vector_type(8))) float v8f;

__device__ __forceinline__ v2f ldg2(const float* p) {
  return *reinterpret_cast<const v2f*>(p);   // global_load_b64 (8B aligned)
}

// ---------------------------------------------------------------------------
// Kernel 1: segmented means (split_nodes).  One block per batch row.
// Phase A: block scan of is_sep -> per-token segment id in LDS.
// Phase B: each of 256 threads owns 3 H-columns with 8 register accumulators;
// streams the 1024x768 fp32 row fully coalesced (402 MB total => the true
// bottleneck; ~17us at 23.3 TB/s).
// ---------------------------------------------------------------------------
__global__ void __launch_bounds__(256)
k_split_nodes(const int* __restrict__ ids, const float* __restrict__ feat,
              float* __restrict__ F /* [NN*BB, HH] row-major */) {
  const int b = blockIdx.x;
  const int t = threadIdx.x;

  __shared__ int   seg[SS];
  __shared__ int   scan[256];
  __shared__ int   cnt[NN];
  __shared__ float inv[NN];

  if (t < NN) cnt[t] = 0;

  int flags[4];
  int partial = 0;
#pragma unroll
  for (int i = 0; i < 4; ++i) {
    const int s = t * 4 + i;
    flags[i] = (ids[(size_t)b * SS + s] == SEP) ? 1 : 0;
    partial += flags[i];
  }
  scan[t] = partial;
  __syncthreads();
  for (int off = 1; off < 256; off <<= 1) {   // Hillis-Steele inclusive scan
    const int self = scan[t];
    const int add  = (t >= off) ? scan[t - off] : 0;
    __syncthreads();
    scan[t] = self + add;
    __syncthreads();
  }
  int run = scan[t] - partial;
#pragma unroll
  for (int i = 0; i < 4; ++i) {
    const int s = t * 4 + i;
    run += flags[i];
    const int sv = (!flags[i] && run < NN) ? run : 0xFF;
    seg[s] = sv;
    if (sv != 0xFF) atomicAdd(&cnt[sv], 1);
  }
  __syncthreads();
  if (t < NN) inv[t] = 1.0f / fmaxf((float)cnt[t], 1.0f);
  __syncthreads();

  float acc[NN][3];
#pragma unroll
  for (int n = 0; n < NN; ++n)
#pragma unroll
    for (int c = 0; c < 3; ++c) acc[n][c] = 0.0f;

  const float* fb = feat + (size_t)b * SS * HH;
  for (int s = 0; s < SS; ++s) {
    const float* row = fb + (size_t)s * HH;
    if ((s & 3) == 0 && s + 4 < SS) {
      __builtin_prefetch(fb + (size_t)(s + 4) * HH + t, 0, 0);  // global_prefetch_b8
    }
    const int sv = seg[s];           // block-uniform -> uniform branch
    if (sv < NN) {
      acc[sv][0] += row[t];
      acc[sv][1] += row[t + 256];
      acc[sv][2] += row[t + 512];
    }
  }

#pragma unroll
  for (int n = 0; n < NN; ++n) {
    float* o = F + ((size_t)n * BB + b) * HH;
    const float s = inv[n];
    o[t]       = acc[n][0] * s;
    o[t + 256] = acc[n][1] * s;
    o[t + 512] = acc[n][2] * s;
  }
}

// ---------------------------------------------------------------------------
// fp32 WMMA GEMM, fully compile-time shapes & leading dimensions.
//   C[M, NB] = op(A[M, KLOOP] @ B[KB, NB] + bias)
// One wave32 computes TWO 16x16 C tiles (rows mp*32..mp*32+31) that share the
// same B fragments: halves B traffic, 2 WMMAs per k-step.
// Guard-free main loop:
//   * KLOOP % 4 == 0 and A's columns KB..KLOOP-1 are ZERO (padded buffers).
//   * Main loop (k0 < KMAIN) needs no B row clamp at all; only the <=3 tail
//     steps clamp with v_min (garbage B values annihilated by zero A).
//   * Column overflow (col >= NB) is address-clamped for loads and masked at
//     the epilogue, preserving the zero-pad invariant for intermediates.
// Lane mapping (ISA 7.12.2, fp32):
//   A: lane L -> A[M=L%16][K = k0 + 2*(L/16) + j], j=0,1  (K-contiguous pair)
//   B: lane L -> B[K = k0 + 2*(L/16) + j][N=L%16]
//   C: lane L, vgpr r -> C[M = r + 8*(L/16)][N = L%16]
// ---------------------------------------------------------------------------
template <int KLOOP, int KB, int NB, int LDA, int LDB, int LDC,
          bool RELU, bool BIAS, bool COMPACT>
__global__ void __launch_bounds__(128)
k_gemm_f32_wmma(const float* __restrict__ A, const float* __restrict__ B,
                const float* __restrict__ bias, float* __restrict__ C) {
  static_assert(KLOOP % 4 == 0, "K loop must be a multiple of the WMMA K=4");
  constexpr int NT = (NB + 15) / 16;                 // N tiles
  // largest k0 such that k0+3 < KB, then +4 => exclusive clamp-free bound
  constexpr int KMAIN = ((KB - 3) & ~3) + 4 > KLOOP ? KLOOP : ((KB - 3) & ~3) + 4;

  const int lane = threadIdx.x & 31;
  const int wave = threadIdx.x >> 5;
  const int tile = blockIdx.x * 4 + wave;            // exact grid, no bounds chk
  const int mp   = tile / NT;                        // M tile-pair index
  const int nt   = tile - mp * NT;

  const int hl   = lane >> 4;
  const int lq   = lane & 15;
  const int row0 = mp * 32 + lq;                     // first tile's A row
  const int col  = nt * 16 + lq;                     // may exceed NB-1
  const int colc = (col < NB) ? col : (NB - 1);      // clamped load column

  const float* Arow0 = A + (size_t)row0 * LDA;
  const float* Arow1 = Arow0 + (size_t)16 * LDA;
  const float* Bcol  = B + colc;
  const int    kbase = 2 * hl;

  v8f acc0 = {};
  v8f acc1 = {};

#pragma unroll 8
  for (int k0 = 0; k0 < KMAIN; k0 += 4) {            // clamp-free main body
    const int kk = k0 + kbase;
    const v2f a0 = ldg2(Arow0 + kk);                 // global_load_b64
    const v2f a1 = ldg2(Arow1 + kk);
    v2f bv;
    bv[0] = Bcol[(size_t)kk * LDB];                  // compile-time stride
    bv[1] = Bcol[(size_t)(kk + 1) * LDB];
    acc0 = __builtin_amdgcn_wmma_f32_16x16x4_f32(
        false, a0, false, bv, (short)0, acc0, false, false);
    acc1 = __builtin_amdgcn_wmma_f32_16x16x4_f32(
        false, a1, false, bv, (short)0, acc1, false, false);
  }
#pragma unroll
  for (int k0 = KMAIN; k0 < KLOOP; k0 += 4) {        // <=3 clamped tail steps
    const int kk  = k0 + kbase;
    const int kb0 = (kk     < KB) ? kk     : (KB - 1);
    const int kb1 = (kk + 1 < KB) ? kk + 1 : (KB - 1);
    const v2f a0 = ldg2(Arow0 + kk);
    const v2f a1 = ldg2(Arow1 + kk);
    v2f bv;
    bv[0] = Bcol[(size_t)kb0 * LDB];
    bv[1] = Bcol[(size_t)kb1 * LDB];
    acc0 = __builtin_amdgcn_wmma_f32_16x16x4_f32(
        false, a0, false, bv, (short)0, acc0, false, false);
    acc1 = __builtin_amdgcn_wmma_f32_16x16x4_f32(
        false, a1, false, bv, (short)0, acc1, false, false);
  }

  float bval = 0.0f;
  if (BIAS) bval = bias[colc];
  const bool colok = (col < NB);

#pragma unroll
  for (int r = 0; r < 8; ++r) {
    const int rr0 = mp * 32 + r + 8 * hl;
    const int rr1 = rr0 + 16;
    float v0 = acc0[r] + bval;
    float v1 = acc1[r] + bval;
    if (RELU) { v0 = fmaxf(v0, 0.0f); v1 = fmaxf(v1, 0.0f); }
    if (COMPACT) {
      if (colok) {
        C[(size_t)rr0 * LDC + col] = v0;             // compact final output
        C[(size_t)rr1 * LDC + col] = v1;
      }
    } else {
      C[(size_t)rr0 * LDC + col] = colok ? v0 : 0.0f;  // keep zero-pad invariant
      C[(size_t)rr1 * LDC + col] = colok ? v1 : 0.0f;
    }
  }
}

// ---------------------------------------------------------------------------
// H1[(i*8+j)*BB+b, c] = relu(P[i*BB+b, c] - P[j*BB+b, c] + b1[c]), 160-wide
// padded output with zeroed pad columns (zero-pad invariant for next GEMM).
// Uses linearity: diff @ W1 == P[i] - P[j] with P = F @ W1.
// ---------------------------------------------------------------------------
__global__ void __launch_bounds__(256)
k_pairwise_relu(const float* __restrict__ P, const float* __restrict__ b1,
                float* __restrict__ H1) {
  const int idx = blockIdx.x * blockDim.x + threadIdx.x;
  if (idx >= NN * NN * BB * CP) return;
  const int c  = idx % CP;
  const int rb = idx / CP;           // (i*8+j)*BB + b
  const int b  = rb % BB;
  const int ij = rb / BB;
  const int i  = ij / NN;
  const int j  = ij - i * NN;
  float v = 0.0f;
  if (c < CC) {
    v = fmaxf(P[((size_t)i * BB + b) * CP + c]
            - P[((size_t)j * BB + b) * CP + c] + b1[c], 0.0f);
  }
  H1[idx] = v;
}

// ---------------------------------------------------------------------------
// Launch. Inputs (setup_inputs order):
//   0 output_ids [B,S] i32 | 1 features [B,S,H] f32 | 2 W1 [H,C] | 3 b1 [C]
//   4 Wm [C,C] | 5 bm [C] | 6 W2 [C,C] | 7 b2 [C]
// Output: [N,N,B,C] f32 (8*8*128*150).
// Workspace (bytes): F 3,145,728 | P 655,360 | H1 5,242,880 | H2 5,242,880
//                    total ~14.3 MB.
// ---------------------------------------------------------------------------
extern "C" void kernel_launch(void* const* d_in, const int* in_sizes, int n_in,
                              void* d_out, int out_size, void* d_ws, size_t ws_size,
                              hipStream_t stream) {
  (void)in_sizes; (void)n_in; (void)out_size; (void)ws_size;

  const int*   ids  = (const int*)d_in[0];
  const float* feat = (const float*)d_in[1];
  const float* W1   = (const float*)d_in[2];
  const float* b1   = (const float*)d_in[3];
  const float* Wm   = (const float*)d_in[4];
  const float* bm   = (const float*)d_in[5];
  const float* W2   = (const float*)d_in[6];
  const float* b2   = (const float*)d_in[7];
  float* out = (float*)d_out;

  char* ws = (char*)d_ws;
  float* F  = (float*)(ws);
  float* P  = (float*)(ws + (size_t)NN * BB * HH * 4);
  float* H1 = (float*)(ws + (size_t)NN * BB * HH * 4 + (size_t)NN * BB * CP * 4);
  float* H2 = (float*)(ws + (size_t)NN * BB * HH * 4 + (size_t)NN * BB * CP * 4
                          + (size_t)NN * NN * BB * CP * 4);

  // 1) segmented means -> F [1024, 768]
  k_split_nodes<<<BB, 256, 0, stream>>>(ids, feat, F);

  // 2) P = F @ W1   M=1024 K=768 N=150 -> padded [1024,160]
  //    (1024/32) * 10 = 320 tile-pairs, 4 waves/block -> 80 blocks
  k_gemm_f32_wmma<HH, HH, CC, HH, CC, CP, false, false, false>
      <<<80, 128, 0, stream>>>(F, W1, nullptr, P);

  // 3) H1 = relu(P[i] - P[j] + b1) -> padded [8192,160], pad cols zeroed
  {
    const int total = NN * NN * BB * CP;          // 1,310,720
    k_pairwise_relu<<<(total + 255) / 256, 256, 0, stream>>>(P, b1, H1);
  }

  // 4) H2 = relu(H1 @ Wm + bm)  M=8192 KLOOP=160(KB=150) N=150 -> [8192,160]
  //    (8192/32) * 10 = 2560 tile-pairs -> 640 blocks
  k_gemm_f32_wmma<CP, CC, CC, CP, CC, CP, true, true, false>
      <<<640, 128, 0, stream>>>(H1, Wm, bm, H2);

  // 5) out = H2 @ W2 + b2  -> compact [8,8,128,150] == output layout
  k_gemm_f32_wmma<CP, CC, CC, CP, CC, CC, false, true, true>
      <<<640, 128, 0, stream>>>(H2, W2, b2, out);
}